// Network_25116968747328
// MI455X (gfx1250) — compile-verified
//
#include <hip/hip_runtime.h>
#include <hip/hip_bf16.h>
#include <math.h>

// ---------------------------------------------------------------------------
// Problem constants (match the reference)
// ---------------------------------------------------------------------------
static constexpr int kB = 8192;   // recurrence length
static constexpr int kD = 2048;   // hidden / feature dim
static constexpr int kE = 128;    // embed size
static constexpr int kTD = 3 * kD; // 6144

typedef __attribute__((ext_vector_type(2))) float v2f;
typedef __attribute__((ext_vector_type(8))) float v8f;

// V_WMMA_F32_16X16X4_F32: full-precision WMMA. 8-arg form per probe:
// (neg_a, A[v2f], neg_b, B[v2f], c_mod, C[v8f], reuse_a, reuse_b)
__device__ inline v8f wmma_k4(v2f a, v2f b, v8f c) {
  return __builtin_amdgcn_wmma_f32_16x16x4_f32(false, a, false, b, (short)0, c,
                                               false, false);
}

// ---------------------------------------------------------------------------
// Kernel 1: gx_all[B,3D] = x @ w_ih^T + b_ih    (206 GFLOP, WMMA f32)
// Grid: (3D/64, B/16) blocks of 128 threads (4 waves, one 16x16 N-tile each).
// A fragment (16x4 f32): lanes 0-15 M=0..15 / K=0,1; lanes 16-31 M / K=2,3.
// B fragment (4x16):     lanes 0-15 N / K=0,1;       lanes 16-31 N / K=2,3.
// ---------------------------------------------------------------------------
__global__ __launch_bounds__(128) void gx_gemm(const float* __restrict__ x,
                                               const float* __restrict__ w_ih,
                                               const float* __restrict__ b_ih,
                                               float* __restrict__ gx) {
  const int lane = threadIdx.x & 31;
  const int wv   = threadIdx.x >> 5;            // 0..3
  const int hi   = lane >> 4;                   // K-half select
  const int l4   = lane & 15;
  const int tileN = blockIdx.x * 4 + wv;        // 0..383
  const int tileM = blockIdx.y;                 // 0..511
  const int arow = tileM * 16 + l4;             // A: M = l4 for both halves
  const int col  = tileN * 16 + l4;             // B/C: N = l4 for both halves

  const float2* __restrict__ xr = (const float2*)(x + (size_t)arow * kD);
  const float2* __restrict__ wr = (const float2*)(w_ih + (size_t)col * kD);

  v8f acc = {};
  for (int k0 = 0; k0 < kD; k0 += 4) {
    const int kb = (k0 + 2 * hi) >> 1;          // float2 index (K pair)
    float2 af = xr[kb];
    float2 bf = wr[kb];                         // B[k,col] = w_ih[col,k]
    v2f a = {af.x, af.y};
    v2f b = {bf.x, bf.y};
    acc = wmma_k4(a, b, acc);
  }
  const float bias = b_ih[col];
#pragma unroll
  for (int v = 0; v < 8; ++v) {
    const int r = tileM * 16 + v + 8 * hi;      // C/D layout: M = v (+8 upper)
    gx[(size_t)r * kTD + col] = acc[v] + bias;
  }
}

// ---------------------------------------------------------------------------
// Kernel 2: persistent GRU scan. 256 WGs x 768 threads (24 waves).
// WG g owns hidden indices j = g*8 .. g*8+7, i.e. w_hh rows {j, D+j, 2D+j}
// (24 rows x 2048 f32 = 192 KB) resident in LDS for all 8192 steps.
// Per step: stage h (8 KB) from L2 into LDS, 24 wave-parallel dot products,
// gate math for 8 hidden units, double-buffered h publish, device barrier.
// Assumes all 256 WGs are co-resident (1 WG/WGP, LDS-limited) — persistent-
// kernel requirement on MI455X.
// ---------------------------------------------------------------------------
__global__ __launch_bounds__(768, 1) void gru_scan(
    const float* __restrict__ gx, const float* __restrict__ w_hh,
    const float* __restrict__ b_hh, float* __restrict__ hg,  // [2, D] ping-pong
    float* __restrict__ Hout,                                 // [B, D]
    unsigned* __restrict__ cnt) {
  extern __shared__ float smem[];
  float* wrows = smem;               // 24*2048
  float* hbuf  = smem + 24 * kD;     // 2048
  float* ghb   = hbuf + kD;          // 24 (+pad)

  const int tid  = threadIdx.x;
  const int wg   = blockIdx.x;       // 0..255
  const int wv   = tid >> 5;         // 0..23  (gate*8 + jj)
  const int lane = tid & 31;
  const int grow = (wv >> 3) * kD + wg * 8 + (wv & 7);

  // One-time: stage this WG's 24 w_hh rows into LDS.
  for (int idx = tid; idx < 24 * kD; idx += 768) {
    const int w = idx >> 11, k = idx & (kD - 1);
    const int gr = (w >> 3) * kD + wg * 8 + (w & 7);
    wrows[idx] = w_hh[(size_t)gr * kD + k];
  }
  __syncthreads();
  const float bh = b_hh[grow];

  for (int t = 0; t < kB; ++t) {
    const float* __restrict__ rbuf = hg + (size_t)(t & 1) * kD;
    float* __restrict__ wbuf       = hg + (size_t)((t + 1) & 1) * kD;

    for (int k = tid; k < kD; k += 768)
      hbuf[k] = (t == 0) ? 0.0f : rbuf[k];
    __syncthreads();

    // gh[row] = w_hh[row,:] . h   (one wave per row, LDS-resident weights)
    const float* __restrict__ wr = wrows + wv * kD;
    float acc = 0.0f;
    for (int k = lane; k < kD; k += 32) acc += wr[k] * hbuf[k];
#pragma unroll
    for (int m = 16; m; m >>= 1) acc += __shfl_xor(acc, m, 32);
    if (lane == 0) ghb[wv] = acc + bh;
    __syncthreads();

    if (tid < 8) {
      const int jl = wg * 8 + tid;
      const float* __restrict__ gxt = gx + (size_t)t * kTD;
      __builtin_prefetch(gx + (size_t)(t + 1) * kTD + jl, 0, 1);
      const float r = 1.0f / (1.0f + expf(-(gxt[jl] + ghb[tid])));
      const float z = 1.0f / (1.0f + expf(-(gxt[kD + jl] + ghb[8 + tid])));
      const float n = tanhf(gxt[2 * kD + jl] + r * ghb[16 + tid]);
      const float hn = (1.0f - z) * n + z * hbuf[jl];
      wbuf[jl] = hn;
      Hout[(size_t)t * kD + jl] = hn;
    }
    __syncthreads();

    // Device-scope split barrier: arrive (release), spin (acquire).
    if (tid == 0) {
      __threadfence();
      atomicAdd(cnt, 1u);
      const unsigned target = (unsigned)(t + 1) * 256u;
      while (__hip_atomic_load(cnt, __ATOMIC_ACQUIRE,
                               __HIP_MEMORY_SCOPE_AGENT) < target) {
        __builtin_amdgcn_s_sleep(1);
      }
    }
    __syncthreads();
  }
}

// ---------------------------------------------------------------------------
// Kernel 3: fused head GEMMs over K=2048 (WMMA f32), one 16-row tile/block.
//   S = relu(H + x) (elementwise, computed in A-fragment layout)
//   logits_p = S@w_p^T + b_p ; logits_n = S@w_n^T + b_n
//   m_out    = sigmoid(H@w_m^T + b_m) ; emb_pre = (x@w_m^T + b_m) + m_out
// 4 waves/block; wave handles 32 of the 128 columns (2 N-tiles) x 4 outputs.
// ---------------------------------------------------------------------------
__global__ __launch_bounds__(128) void heads_gemm(
    const float* __restrict__ H, const float* __restrict__ x,
    const float* __restrict__ w_p, const float* __restrict__ w_n,
    const float* __restrict__ w_m, const float* __restrict__ b_p,
    const float* __restrict__ b_n, const float* __restrict__ b_m,
    float* __restrict__ logits_p, float* __restrict__ logits_n,
    float* __restrict__ m_out, float* __restrict__ emb_pre) {
  const int lane = threadIdx.x & 31;
  const int wv   = threadIdx.x >> 5;
  const int hi   = lane >> 4;
  const int l4   = lane & 15;
  const int tileM = blockIdx.x;                 // 0..511
  const int arow  = tileM * 16 + l4;
  const int col0  = wv * 32 + l4;
  const int col1  = col0 + 16;

  const float2* __restrict__ hr = (const float2*)(H + (size_t)arow * kD);
  const float2* __restrict__ xr = (const float2*)(x + (size_t)arow * kD);
  const float2* __restrict__ p0 = (const float2*)(w_p + (size_t)col0 * kD);
  const float2* __restrict__ p1 = (const float2*)(w_p + (size_t)col1 * kD);
  const float2* __restrict__ n0 = (const float2*)(w_n + (size_t)col0 * kD);
  const float2* __restrict__ n1 = (const float2*)(w_n + (size_t)col1 * kD);
  const float2* __restrict__ m0 = (const float2*)(w_m + (size_t)col0 * kD);
  const float2* __restrict__ m1 = (const float2*)(w_m + (size_t)col1 * kD);

  v8f aP0 = {}, aP1 = {}, aN0 = {}, aN1 = {};
  v8f aM0 = {}, aM1 = {}, aX0 = {}, aX1 = {};

  for (int k0 = 0; k0 < kD; k0 += 4) {
    const int kb = (k0 + 2 * hi) >> 1;
    float2 hf = hr[kb], xf = xr[kb];
    v2f hA = {hf.x, hf.y};
    v2f xA = {xf.x, xf.y};
    v2f sA = {fmaxf(hf.x + xf.x, 0.0f), fmaxf(hf.y + xf.y, 0.0f)};
    float2 bp0 = p0[kb], bp1 = p1[kb];
    float2 bn0 = n0[kb], bn1 = n1[kb];
    float2 bm0 = m0[kb], bm1 = m1[kb];
    aP0 = wmma_k4(sA, (v2f){bp0.x, bp0.y}, aP0);
    aP1 = wmma_k4(sA, (v2f){bp1.x, bp1.y}, aP1);
    aN0 = wmma_k4(sA, (v2f){bn0.x, bn0.y}, aN0);
    aN1 = wmma_k4(sA, (v2f){bn1.x, bn1.y}, aN1);
    aM0 = wmma_k4(hA, (v2f){bm0.x, bm0.y}, aM0);
    aM1 = wmma_k4(hA, (v2f){bm1.x, bm1.y}, aM1);
    aX0 = wmma_k4(xA, (v2f){bm0.x, bm0.y}, aX0);
    aX1 = wmma_k4(xA, (v2f){bm1.x, bm1.y}, aX1);
  }

  const float bp_0 = b_p[col0], bp_1 = b_p[col1];
  const float bn_0 = b_n[col0], bn_1 = b_n[col1];
  const float bm_0 = b_m[col0], bm_1 = b_m[col1];
#pragma unroll
  for (int v = 0; v < 8; ++v) {
    const int r = tileM * 16 + v + 8 * hi;
    const size_t o0 = (size_t)r * kE + col0;
    const size_t o1 = (size_t)r * kE + col1;
    logits_p[o0] = aP0[v] + bp_0;
    logits_p[o1] = aP1[v] + bp_1;
    logits_n[o0] = aN0[v] + bn_0;
    logits_n[o1] = aN1[v] + bn_1;
    const float mv0 = 1.0f / (1.0f + expf(-(aM0[v] + bm_0)));
    const float mv1 = 1.0f / (1.0f + expf(-(aM1[v] + bm_1)));
    m_out[o0] = mv0;
    m_out[o1] = mv1;
    emb_pre[o0] = (aX0[v] + bm_0) + mv0;
    emb_pre[o1] = (aX1[v] + bm_1) + mv1;
  }
}

// ---------------------------------------------------------------------------
// Kernel 4: per-row log-softmax + Gumbel argmax + normalize + output packing.
// One block of 128 threads per row (4 waves).
// ---------------------------------------------------------------------------
__device__ inline float blkmax(float v, float* sred) {
#pragma unroll
  for (int m = 16; m; m >>= 1) v = fmaxf(v, __shfl_xor(v, m, 32));
  if ((threadIdx.x & 31) == 0) sred[threadIdx.x >> 5] = v;
  __syncthreads();
  v = fmaxf(fmaxf(sred[0], sred[1]), fmaxf(sred[2], sred[3]));
  __syncthreads();
  return v;
}
__device__ inline float blksum(float v, float* sred) {
#pragma unroll
  for (int m = 16; m; m >>= 1) v += __shfl_xor(v, m, 32);
  if ((threadIdx.x & 31) == 0) sred[threadIdx.x >> 5] = v;
  __syncthreads();
  v = sred[0] + sred[1] + sred[2] + sred[3];
  __syncthreads();
  return v;
}
__device__ inline int blkmin_i(int v, int* sidx) {
#pragma unroll
  for (int m = 16; m; m >>= 1) v = min(v, __shfl_xor(v, m, 32));
  if ((threadIdx.x & 31) == 0) sidx[threadIdx.x >> 5] = v;
  __syncthreads();
  v = min(min(sidx[0], sidx[1]), min(sidx[2], sidx[3]));
  __syncthreads();
  return v;
}

__global__ __launch_bounds__(128) void finalize(
    const float* __restrict__ logits_p, const float* __restrict__ logits_n,
    const float* __restrict__ m_out, const float* __restrict__ emb_pre,
    const float* __restrict__ g_p, const float* __restrict__ g_n,
    float* __restrict__ out) {
  __shared__ float sred[4];
  __shared__ int sidx[4];
  __shared__ float spv[2];
  const int b = blockIdx.x;
  const int e = threadIdx.x;
  const size_t o = (size_t)b * kE + e;
  const size_t BE = (size_t)kB * kE;

  // p-head
  float lp = logits_p[o];
  float mx = blkmax(lp, sred);
  float se = blksum(expf(lp - mx), sred);
  float logp = lp - mx - logf(se);
  float y = logp + g_p[o];
  float ym = blkmax(y, sred);
  int ap = blkmin_i((y == ym) ? e : kE, sidx);
  if (e == ap) spv[0] = logp;

  // n-head
  float ln_ = logits_n[o];
  float mx2 = blkmax(ln_, sred);
  float se2 = blksum(expf(ln_ - mx2), sred);
  float logn = ln_ - mx2 - logf(se2);
  float y2 = logn + g_n[o];
  float ym2 = blkmax(y2, sred);
  int an = blkmin_i((y2 == ym2) ? e : kE, sidx);
  if (e == an) spv[1] = logn;
  __syncthreads();
  const float probp = spv[0];
  const float probn = spv[1];

  // normalize embedding and rnn_output
  const float mv = m_out[o];
  const float ev = emb_pre[o];
  const float ne = sqrtf(blksum(ev * ev, sred));
  const float nm = sqrtf(blksum(mv * mv, sred));
  out[o] = ev / fmaxf(ne, 1e-12f);
  out[BE + o] = mv / fmaxf(nm, 1e-12f);
  if (e == 0) {
    out[2 * BE + b] = probp;
    out[2 * BE + kB + b] = probn;
    out[2 * BE + 2 * kB + b] = (float)(ap + 1) * (1.0f / (float)kE);
    out[2 * BE + 3 * kB + b] = (float)(an + 1) * (1.0f / (float)kE);
  }
}

// ---------------------------------------------------------------------------
// Host-side launch (graph-capture safe: only async ops on `stream`).
// ---------------------------------------------------------------------------
extern "C" void kernel_launch(void* const* d_in, const int* in_sizes, int n_in,
                              void* d_out, int out_size, void* d_ws,
                              size_t ws_size, hipStream_t stream) {
  const float* x    = (const float*)d_in[0];
  const float* w_ih = (const float*)d_in[1];
  const float* w_hh = (const float*)d_in[2];
  const float* b_ih = (const float*)d_in[3];
  const float* b_hh = (const float*)d_in[4];
  const float* w_p  = (const float*)d_in[5];
  const float* b_p  = (const float*)d_in[6];
  const float* w_n  = (const float*)d_in[7];
  const float* b_n  = (const float*)d_in[8];
  const float* w_m  = (const float*)d_in[9];
  const float* b_m  = (const float*)d_in[10];
  const float* g_p  = (const float*)d_in[11];
  const float* g_n  = (const float*)d_in[12];
  float* out = (float*)d_out;
  float* ws  = (float*)d_ws;

  // Workspace layout (floats): ~285 MB total.
  const size_t oGX  = 0;                                  // [B,3D]
  const size_t oH   = oGX + (size_t)kB * kTD;             // [B,D]
  const size_t oHG  = oH + (size_t)kB * kD;               // [2,D] ping-pong h
  const size_t oLP  = oHG + 2 * (size_t)kD;               // [B,E]
  const size_t oLN  = oLP + (size_t)kB * kE;              // [B,E]
  const size_t oMO  = oLN + (size_t)kB * kE;              // [B,E]
  const size_t oEP  = oMO + (size_t)kB * kE;              // [B,E]
  const size_t oCNT = oEP + (size_t)kB * kE;              // 1 (uint)
  unsigned* cnt = (unsigned*)(ws + oCNT);

  hipMemsetAsync(cnt, 0, sizeof(unsigned), stream);

  // 1) gx = x @ w_ih^T + b_ih
  dim3 g1(kTD / 64, kB / 16);
  gx_gemm<<<g1, 128, 0, stream>>>(x, w_ih, b_ih, ws + oGX);

  // 2) persistent GRU scan (204 KB dynamic LDS per WG)
  const size_t ldsBytes = (size_t)(24 * kD + kD + 32) * sizeof(float);
  hipFuncSetAttribute(reinterpret_cast<const void*>(gru_scan),
                      hipFuncAttributeMaxDynamicSharedMemorySize,
                      (int)ldsBytes);
  gru_scan<<<256, 768, ldsBytes, stream>>>(ws + oGX, w_hh, b_hh, ws + oHG,
                                           ws + oH, cnt);

  // 3) fused head GEMMs
  heads_gemm<<<kB / 16, 128, 0, stream>>>(ws + oH, x, w_p, w_n, w_m, b_p, b_n,
                                          b_m, ws + oLP, ws + oLN, ws + oMO,
                                          ws + oEP);

  // 4) softmax / argmax / normalize / pack outputs
  finalize<<<kB, 128, 0, stream>>>(ws + oLP, ws + oLN, ws + oMO, ws + oEP,
                                   g_p, g_n, out);
}